// GaussianRasterizer_10548439679665
// MI455X (gfx1250) — compile-verified
//
#include <hip/hip_runtime.h>
#include <hip/hip_bf16.h>
#include <math.h>

typedef __attribute__((ext_vector_type(2))) float v2f;
typedef __attribute__((ext_vector_type(8))) float v8f;

#define NP 1024
#define IMH 160
#define IMW 160
#define REC 12           // floats per gaussian record (3 x float4)
#define NPIX (IMH*IMW)

// ---------------------------------------------------------------------------
// Kernel A: per-gaussian preprocessing.
// Produces: radii (int, into d_out tail), depth key, and a 12-float record
//   [px, py, conic_a, conic_b, conic_c, opacity, r, g, b, 0,0,0]
// Invalid gaussians get an all-zero record (=> alpha==0 in the blend, which
// matches the reference's ~valid mask) and key=+inf (sort to the back).
// ---------------------------------------------------------------------------
__global__ void gs_preprocess(const float* __restrict__ means3D,
                              const float* __restrict__ colors,
                              const float* __restrict__ opac,
                              const float* __restrict__ scales,
                              const float* __restrict__ rot,
                              const float* __restrict__ viewm,
                              const float* __restrict__ projm,
                              float4* __restrict__ wsU,       // NP * 3 float4
                              float* __restrict__ wsKey,
                              int* __restrict__ radii_out) {
    int i = blockIdx.x * blockDim.x + threadIdx.x;
    if (i >= NP) return;

    const float fx = 160.0f, fy = 160.0f;      // W/(2*tanfovx), H/(2*tanfovy)
    const float limx = 0.65f, limy = 0.65f;    // 1.3*tanfov

    float hom[4] = { means3D[i*3+0], means3D[i*3+1], means3D[i*3+2], 1.0f };

    float t[4];
    #pragma unroll
    for (int r = 0; r < 4; ++r)
        t[r] = viewm[r*4+0]*hom[0] + viewm[r*4+1]*hom[1] +
               viewm[r*4+2]*hom[2] + viewm[r*4+3]*hom[3];
    float tz = t[2];
    bool depth_ok = tz > 0.2f;

    float ph[4];
    #pragma unroll
    for (int r = 0; r < 4; ++r)
        ph[r] = projm[r*4+0]*hom[0] + projm[r*4+1]*hom[1] +
                projm[r*4+2]*hom[2] + projm[r*4+3]*hom[3];
    float pwv = 1.0f / (ph[3] + 1e-7f);
    float p0 = ph[0]*pwv, p1 = ph[1]*pwv;
    float px = ((p0 + 1.0f) * (float)IMW - 1.0f) * 0.5f;
    float py = ((p1 + 1.0f) * (float)IMH - 1.0f) * 0.5f;

    // quaternion -> rotation
    float qr = rot[i*4+0], qx = rot[i*4+1], qy = rot[i*4+2], qz = rot[i*4+3];
    float R[3][3];
    R[0][0] = 1.f - 2.f*(qy*qy + qz*qz); R[0][1] = 2.f*(qx*qy - qr*qz); R[0][2] = 2.f*(qx*qz + qr*qy);
    R[1][0] = 2.f*(qx*qy + qr*qz); R[1][1] = 1.f - 2.f*(qx*qx + qz*qz); R[1][2] = 2.f*(qy*qz - qr*qx);
    R[2][0] = 2.f*(qx*qz - qr*qy); R[2][1] = 2.f*(qy*qz + qr*qx); R[2][2] = 1.f - 2.f*(qx*qx + qy*qy);

    float s0 = scales[i*3+0], s1 = scales[i*3+1], s2 = scales[i*3+2];
    float M[3][3];
    #pragma unroll
    for (int r = 0; r < 3; ++r) { M[r][0] = R[r][0]*s0; M[r][1] = R[r][1]*s1; M[r][2] = R[r][2]*s2; }
    float C3[3][3];
    #pragma unroll
    for (int r = 0; r < 3; ++r)
        #pragma unroll
        for (int c = 0; c < 3; ++c)
            C3[r][c] = M[r][0]*M[c][0] + M[r][1]*M[c][1] + M[r][2]*M[c][2];

    float itz = 1.0f / tz;
    float txz = fminf(fmaxf(t[0]*itz, -limx), limx) * tz;
    float tyz = fminf(fmaxf(t[1]*itz, -limy), limy) * tz;
    float J[2][3];
    J[0][0] = fx*itz; J[0][1] = 0.f;    J[0][2] = -fx*txz*itz*itz;
    J[1][0] = 0.f;    J[1][1] = fy*itz; J[1][2] = -fy*tyz*itz*itz;

    // T = J @ Wm ; Wm[j][k] = viewm[j*4+k]
    float T[2][3];
    #pragma unroll
    for (int r = 0; r < 2; ++r)
        #pragma unroll
        for (int k = 0; k < 3; ++k)
            T[r][k] = J[r][0]*viewm[0*4+k] + J[r][1]*viewm[1*4+k] + J[r][2]*viewm[2*4+k];

    float tmp[2][3];
    #pragma unroll
    for (int r = 0; r < 2; ++r)
        #pragma unroll
        for (int k = 0; k < 3; ++k)
            tmp[r][k] = T[r][0]*C3[0][k] + T[r][1]*C3[1][k] + T[r][2]*C3[2][k];
    float cov00 = tmp[0][0]*T[0][0] + tmp[0][1]*T[0][1] + tmp[0][2]*T[0][2];
    float cov01 = tmp[0][0]*T[1][0] + tmp[0][1]*T[1][1] + tmp[0][2]*T[1][2];
    float cov11 = tmp[1][0]*T[1][0] + tmp[1][1]*T[1][1] + tmp[1][2]*T[1][2];

    float a = cov00 + 0.3f, c = cov11 + 0.3f, b = cov01;
    float det = a*c - b*b;
    bool det_ok = det > 0.0f;
    float inv_det = det_ok ? (1.0f / det) : 0.0f;
    float conic_a =  c*inv_det;
    float conic_b = -b*inv_det;
    float conic_c =  a*inv_det;
    float mid = 0.5f*(a + c);
    float lam1 = mid + sqrtf(fmaxf(0.1f, mid*mid - det));
    float radius = ceilf(3.0f * sqrtf(lam1));

    bool valid = depth_ok && det_ok;
    radii_out[i] = valid ? (int)radius : 0;
    wsKey[i] = valid ? tz : __int_as_float(0x7f800000);

    float4 r0, r1, r2;
    if (valid) {
        r0 = make_float4(px, py, conic_a, conic_b);
        r1 = make_float4(conic_c, opac[i], colors[i*3+0], colors[i*3+1]);
        r2 = make_float4(colors[i*3+2], 0.f, 0.f, 0.f);
    } else {
        r0 = make_float4(0.f, 0.f, 0.f, 0.f);
        r1 = r0; r2 = r0;
    }
    wsU[i*3 + 0] = r0;
    wsU[i*3 + 1] = r1;
    wsU[i*3 + 2] = r2;
}

// ---------------------------------------------------------------------------
// Kernel B: stable ascending rank-sort by depth key (O(P^2), P=1024 -> 1M cmp)
// Scatters records (as float4s) into sorted order.
// ---------------------------------------------------------------------------
__global__ void gs_sort(const float* __restrict__ wsKey,
                        const float4* __restrict__ wsU,
                        float4* __restrict__ wsS) {
    __shared__ float skey[NP];
    int tid = threadIdx.x;
    for (int j = tid; j < NP; j += blockDim.x) skey[j] = wsKey[j];
    __syncthreads();

    int i = blockIdx.x * blockDim.x + tid;
    if (i >= NP) return;
    float ki = skey[i];
    int rank = 0;
    for (int j = 0; j < NP; ++j) {
        float kj = skey[j];
        rank += (kj < ki) || ((kj == ki) && (j < i));
    }
    float4 r0 = wsU[i*3 + 0];
    float4 r1 = wsU[i*3 + 1];
    float4 r2 = wsU[i*3 + 2];
    wsS[rank*3 + 0] = r0;
    wsS[rank*3 + 1] = r1;
    wsS[rank*3 + 2] = r2;
}

// ---------------------------------------------------------------------------
// Kernel C: tiled rasterization. One wave32 per 16-pixel row tile.
// Per 16-gaussian chunk: power(g,x) = c0 + c1*x + c2*x^2 evaluated as a
// (16x4)@(4x16) f32 WMMA; then in-register front-to-back blend (8 gaussians
// per half-wave, merged via the associative compositing rule with shfl_xor).
// ---------------------------------------------------------------------------
__global__ void __launch_bounds__(256)
gs_raster(const float4* __restrict__ wsS,
          const float* __restrict__ bg,
          float* __restrict__ out) {
    __shared__ float sdat[NP * REC];   // 48 KB sorted records
    int tid = threadIdx.x;

    // prefetch this block's view of the sorted records (global_prefetch_b8),
    // then stage via B128 global loads -> B128 LDS stores.
    __builtin_prefetch((const char*)wsS + (size_t)tid * 192, 0, 0);
    {
        float4* sdat4 = (float4*)sdat;
        for (int j = tid; j < NP * REC / 4; j += blockDim.x)
            sdat4[j] = wsS[j];
    }
    __syncthreads();

    unsigned lane = tid & 31u;
    unsigned wave = tid >> 5;                 // 8 waves / block
    unsigned tileId = blockIdx.x * 8u + wave; // 1600 tiles: 10 x-tiles * 160 rows
    int x0 = (int)(tileId % 10u) * 16;
    int yt = (int)(tileId / 10u);

    bool loHalf = lane < 16u;
    float xn = (float)(lane & 15u);

    // B fragment: rows K=0..3 of pixel-feature matrix [1, x, x^2, 0]
    v2f bfrag;
    bfrag.x = loHalf ? 1.0f : xn * xn;   // K=0 / K=2
    bfrag.y = loHalf ? xn   : 0.0f;      // K=1 / K=3

    float CR = 0.f, CG = 0.f, CB = 0.f, TT = 1.f;  // per-pixel state (lanes 0-15)
    float fx0 = (float)x0, fyt = (float)yt;

    for (int chunk = 0; chunk < NP / 16; ++chunk) {
        int g0 = chunk * 16;

        // A fragment: gaussian row M = lane&15, coeffs of [1, x, x^2]
        int gi = (g0 + (int)(lane & 15u)) * REC;
        float gpx = sdat[gi + 0] - fx0;
        float dyv = sdat[gi + 1] - fyt;
        float ca  = sdat[gi + 2];
        float cb2 = sdat[gi + 3];
        float cc  = sdat[gi + 4];
        float c0 = -0.5f*ca*gpx*gpx - 0.5f*cc*dyv*dyv - cb2*gpx*dyv;
        float c1 = ca*gpx + cb2*dyv;
        float c2 = -0.5f*ca;
        v2f afrag;
        afrag.x = loHalf ? c0 : c2;
        afrag.y = loHalf ? c1 : 0.0f;

        v8f cz = {0.f, 0.f, 0.f, 0.f, 0.f, 0.f, 0.f, 0.f};
        v8f dmat = __builtin_amdgcn_wmma_f32_16x16x4_f32(
            false, afrag, false, bfrag, (short)0, cz, false, false);

        // half-wave blend: lanes 0-15 -> gaussians g0+0..7, lanes 16-31 -> g0+8..15
        float Tc = 1.f, pr = 0.f, pg = 0.f, pb = 0.f;
        int gbase = g0 + (loHalf ? 0 : 8);
        #pragma unroll
        for (int r = 0; r < 8; ++r) {
            float pw = dmat[r];
            int g = (gbase + r) * REC;
            float op = sdat[g + 5];
            float al = fminf(0.99f, op * __expf(pw));
            al = (pw > 0.0f || al < (1.0f / 255.0f)) ? 0.0f : al;
            float w = Tc * al;
            pr += w * sdat[g + 6];
            pg += w * sdat[g + 7];
            pb += w * sdat[g + 8];
            Tc *= (1.0f - al);
        }

        // merge halves: (C,T) o (C',T') = (C + T*C', T*T')
        float pr2 = __shfl_xor(pr, 16, 32);
        float pg2 = __shfl_xor(pg, 16, 32);
        float pb2 = __shfl_xor(pb, 16, 32);
        float Tc2 = __shfl_xor(Tc, 16, 32);
        pr += Tc * pr2;  pg += Tc * pg2;  pb += Tc * pb2;  Tc *= Tc2;

        // compose into persistent per-pixel state (meaningful in lanes 0-15)
        CR += TT * pr;  CG += TT * pg;  CB += TT * pb;  TT *= Tc;
    }

    if (loHalf) {
        int x = x0 + (int)lane;
        int pix = yt * IMW + x;
        out[0 * NPIX + pix] = CR + TT * bg[0];
        out[1 * NPIX + pix] = CG + TT * bg[1];
        out[2 * NPIX + pix] = CB + TT * bg[2];
    }
}

// ---------------------------------------------------------------------------
extern "C" void kernel_launch(void* const* d_in, const int* in_sizes, int n_in,
                              void* d_out, int out_size, void* d_ws, size_t ws_size,
                              hipStream_t stream) {
    const float* means3D = (const float*)d_in[0];
    // d_in[1] = means2D (unused)
    const float* colors  = (const float*)d_in[2];
    const float* opac    = (const float*)d_in[3];
    const float* scales  = (const float*)d_in[4];
    const float* rot     = (const float*)d_in[5];
    const float* bg      = (const float*)d_in[6];
    const float* viewm   = (const float*)d_in[7];
    const float* projm   = (const float*)d_in[8];
    // d_in[9] = campos (unused)

    float* out   = (float*)d_out;               // 3*160*160 color
    int*   radii = ((int*)d_out) + 3 * NPIX;    // 1024 int32 radii

    float4* wsU   = (float4*)d_ws;                      // 1024*3 float4 unsorted
    float*  wsKey = (float*)d_ws + NP * REC;            // 1024 depth keys
    float4* wsS   = (float4*)((float*)d_ws + NP * REC + NP); // 1024*3 float4 sorted

    gs_preprocess<<<4, 256, 0, stream>>>(means3D, colors, opac, scales, rot,
                                         viewm, projm, wsU, wsKey, radii);
    gs_sort<<<4, 256, 0, stream>>>(wsKey, wsU, wsS);
    gs_raster<<<200, 256, 0, stream>>>(wsS, bg, out);
}